// GaussianMoments_60748017435174
// MI455X (gfx1250) — compile-verified
//
#include <hip/hip_runtime.h>
#include <hip/hip_bf16.h>

typedef __attribute__((ext_vector_type(2))) float v2f;
typedef __attribute__((ext_vector_type(8))) float v8f;

#define DIMC 32
#define KC   10
#define BC   2048
#define PT   2080   // padded row capacity >= BC + 3*KC, multiple of 32

// compress constants (order 2 / 3 / 4)
#define C2A 0.25f
#define P2  0.5f
#define C2S 0.5f
#define C3A 0.19245008973f
#define P3  0.3333333333f
#define C3S 0.57735026919f
#define C4A 0.15749013123f
#define P4  0.25f
#define C4S 0.62996052494f

// wave-tiled GEMM partial counts
#define NP4  (KC * 32 * 16)     // 5120 m4 waves: 2x4 tile blocks of 64x64 tiles
#define NP3  (KC * 32)          // 320  m3 waves: 2x2 tile blocks of 64x2 tiles
#define NP12 ((KC * 1024) / 32) // 320  m1/m2 waves

#define WMMA_F32(A, B, C) \
    __builtin_amdgcn_wmma_f32_16x16x4_f32(false, (A), false, (B), (short)0, (C), false, false)

__device__ __forceinline__ float compressf(float m, float ca, float pw, float cs) {
    float s = (m >= 0.0f) ? 1.0f : -1.0f;     // sign(sign(m)+0.1)
    return s * (powf(fabsf(m) + ca, pw) - cs);
}

__device__ __forceinline__ float permw3(int a, int b, int c) {
    int t;
    if (a > b) { t = a; a = b; b = t; }
    if (b > c) { t = b; b = c; c = t; }
    if (a > b) { t = a; a = b; b = t; }
    float cnt = 1.0f, pr = 1.0f;
    cnt = (b == a) ? cnt + 1.0f : 1.0f; pr *= cnt;
    cnt = (c == b) ? cnt + 1.0f : 1.0f; pr *= cnt;
    return pr * (1.0f / 6.0f);
}

__device__ __forceinline__ float permw4(int a, int b, int c, int d) {
    int t;
    if (a > b) { t = a; a = b; b = t; }
    if (c > d) { t = c; c = d; d = t; }
    if (a > c) { t = a; a = c; c = t; }
    if (b > d) { t = b; b = d; d = t; }
    if (b > c) { t = b; b = c; c = t; }
    float cnt = 1.0f, pr = 1.0f;
    cnt = (b == a) ? cnt + 1.0f : 1.0f; pr *= cnt;
    cnt = (c == b) ? cnt + 1.0f : 1.0f; pr *= cnt;
    cnt = (d == c) ? cnt + 1.0f : 1.0f; pr *= cnt;
    return pr * (1.0f / 24.0f);
}

// Load one f32 WMMA 16x16x4 fragment (A 16x4 or B 4x16 — same lane pattern):
// lane l: ln = l&15, half = l>>4 ; f.x = src[row s+2*half][colbase+ln], f.y = row+1
__device__ __forceinline__ v2f load_frag(const float* base, int lane, int s,
                                         int colbase, int rowstride) {
    int half = lane >> 4;
    int ln   = lane & 15;
    size_t r0 = (size_t)(s + 2 * half);
    v2f f;
    f.x = base[r0 * rowstride + colbase + ln];
    f.y = base[(r0 + 1) * rowstride + colbase + ln];
    return f;
}

// ---------------- kernels ----------------

__global__ void k_init(float* X, float* Dm, int* order) {
    size_t i  = (size_t)blockIdx.x * blockDim.x + threadIdx.x;
    size_t nX = (size_t)PT * 1024;
    size_t nD = (size_t)PT * DIMC;
    if (i < nX) X[i] = 0.0f;
    if (i < nD) Dm[i] = 0.0f;
    if (i < PT) order[i] = -1;
}

__global__ void k_assign(const float* __restrict__ logits, int* __restrict__ asg) {
    int b = blockIdx.x * blockDim.x + threadIdx.x;
    if (b >= BC) return;
    const float* row = logits + (size_t)b * KC;
    int best = 0; float bv = row[0];
    for (int k = 1; k < KC; ++k) {
        float v = row[k];
        if (v > bv) { bv = v; best = k; }   // first-max tie-break, matches argmax
    }
    asg[b] = best;
}

__global__ void k_meta(const int* __restrict__ asg, int* __restrict__ poff,
                       float* __restrict__ cwn, float* __restrict__ cnt) {
    if (threadIdx.x != 0 || blockIdx.x != 0) return;
    int c[KC];
    for (int k = 0; k < KC; ++k) c[k] = 0;
    for (int b = 0; b < BC; ++b) c[asg[b]]++;
    float tot = 0.0f;
    for (int k = 0; k < KC; ++k) tot += (float)c[k];
    int off = 0;
    for (int k = 0; k < KC; ++k) {
        poff[k] = off;
        off += (c[k] + 3) & ~3;            // pad each cluster to multiple of 4 (WMMA K step)
    }
    poff[KC] = off;
    for (int k = 0; k < KC; ++k) {
        cwn[k] = (float)c[k] / tot;
        cnt[k] = (float)c[k] + 1e-7f;
    }
}

// deterministic scatter: rank of b within its cluster by sample index order
__global__ void k_scatter(const int* __restrict__ asg, const int* __restrict__ poff,
                          int* __restrict__ order) {
    int b = blockIdx.x * blockDim.x + threadIdx.x;
    if (b >= BC) return;
    int k = asg[b];
    int r = 0;
    for (int q = 0; q < b; ++q) r += (asg[q] == k) ? 1 : 0;
    order[poff[k] + r] = b;
}

// build X[p][i*32+j] = d_i*d_j and Dm[p][d] = d_d for each compacted row
__global__ void k_build(const float* __restrict__ emb, const float* __restrict__ cen,
                        const int* __restrict__ order, const int* __restrict__ poff,
                        float* __restrict__ X, float* __restrict__ Dm) {
    int p = blockIdx.x;
    int b = order[p];
    if (b < 0) return;                      // padding row: stays zero from k_init
    __shared__ float sh[DIMC];
    int k = 0;
    for (int q = 0; q < KC; ++q)
        if (p >= poff[q]) k = q;
    if (threadIdx.x < DIMC) {
        float d = emb[(size_t)b * DIMC + threadIdx.x] - cen[(size_t)k * DIMC + threadIdx.x];
        sh[threadIdx.x] = d;
        Dm[(size_t)p * DIMC + threadIdx.x] = d;
    }
    __syncthreads();
    for (int t = threadIdx.x; t < 1024; t += blockDim.x)
        X[(size_t)p * 1024 + t] = sh[t >> 5] * sh[t & 31];
}

// m1 + m2 losses (row-sum reductions over X / Dm), per-wave partials
__global__ void k_m12(const float* __restrict__ X, const float* __restrict__ Dm,
                      const int* __restrict__ poff, const float* __restrict__ cwn,
                      const float* __restrict__ cnt, float* __restrict__ part) {
    int tid = blockIdx.x * blockDim.x + threadIdx.x;   // KC*1024 threads exactly
    int k = tid >> 10, col = tid & 1023;
    int s0 = poff[k], s1 = poff[k + 1];
    float acc = 0.0f;
    for (int p = s0; p < s1; ++p) acc += X[(size_t)p * 1024 + col];
    float m2 = acc / cnt[k];
    int i = col >> 5, j = col & 31;
    float term = 0.0f;
    if (i != j) {                                       // off_diag mask
        float m2c = compressf(m2, C2A, P2, C2S);
        float t2c = compressf(0.0f, C2A, P2, C2S);      // t2 off-diagonal = 0
        float d = m2c - t2c;
        term += 0.5f * (cwn[k] * 0.5f) * d * d;         // 0.5 * cwn * w2(=0.5) * d^2
    }
    if (col < DIMC) {
        float a1 = 0.0f;
        for (int p = s0; p < s1; ++p) a1 += Dm[(size_t)p * DIMC + col];
        float m1 = a1 / cnt[k];
        term += 1.0f * cwn[k] * m1 * m1;                // w1 = 1, t1 = 0
    }
    for (int m = 16; m > 0; m >>= 1) term += __shfl_xor(term, m, 32);
    if ((threadIdx.x & 31) == 0) part[NP4 + NP3 + (tid >> 5)] = term;
}

// m3 = X^T D per cluster (1024x32 GEMM), 2x2 register-tiled, fused order-3 loss
__global__ void k_m3(const float* __restrict__ X, const float* __restrict__ Dm,
                     const int* __restrict__ poff, const float* __restrict__ cwn,
                     float* __restrict__ part) {
    int wave = (blockIdx.x * blockDim.x + threadIdx.x) >> 5;  // NP3 waves exactly
    int lane = threadIdx.x & 31;
    int k   = wave >> 5;            // wave / 32
    int ti0 = (wave & 31) * 2;      // two 16-row tiles of the 1024 dim
    int s0 = poff[k], s1 = poff[k + 1];
    v8f acc[4];
#pragma unroll
    for (int t = 0; t < 4; ++t) acc[t] = (v8f){0.f,0.f,0.f,0.f,0.f,0.f,0.f,0.f};
    for (int s = s0; s < s1; s += 4) {
        v2f a0 = load_frag(X,  lane, s, (ti0 + 0) * 16, 1024);
        v2f a1 = load_frag(X,  lane, s, (ti0 + 1) * 16, 1024);
        v2f b0 = load_frag(Dm, lane, s, 0,  DIMC);
        v2f b1 = load_frag(Dm, lane, s, 16, DIMC);
        acc[0] = WMMA_F32(a0, b0, acc[0]);
        acc[1] = WMMA_F32(a0, b1, acc[1]);
        acc[2] = WMMA_F32(a1, b0, acc[2]);
        acc[3] = WMMA_F32(a1, b1, acc[3]);
    }
    float cw  = cwn[k];
    float t3c = compressf(0.0f, C3A, P3, C3S);
    int n     = lane & 15;
    int mbase = (lane >> 4) * 8;
    float term = 0.0f;
#pragma unroll
    for (int p = 0; p < 2; ++p) {
#pragma unroll
        for (int q = 0; q < 2; ++q) {
            v8f cacc = acc[p * 2 + q];
#pragma unroll
            for (int v = 0; v < 8; ++v) {
                int r = (ti0 + p) * 16 + mbase + v;   // flattened (i,j)
                int i = r >> 5, j = r & 31;
                int l = q * 16 + n;
                float w3  = permw3(i, j, l);
                float m3c = compressf(cacc[v], C3A, P3, C3S);
                float d   = m3c - t3c;
                term += 0.25f * cw * w3 * d * d;
            }
        }
    }
    for (int m = 16; m > 0; m >>= 1) term += __shfl_xor(term, m, 32);
    if (lane == 0) part[NP4 + wave] = term;
}

// m4 = X^T X per cluster (1024x1024 Gram), 2x4 register-tiled, fused order-4 loss
__global__ void k_m4(const float* __restrict__ X, const int* __restrict__ poff,
                     const float* __restrict__ cwn, float* __restrict__ part) {
    int wave = (blockIdx.x * blockDim.x + threadIdx.x) >> 5;  // NP4 waves exactly
    int lane = threadIdx.x & 31;
    int k   = wave >> 9;                 // wave / 512
    int rem = wave & 511;
    int ti0 = (rem >> 4) * 2;            // 2 row-tiles  (M block of 32)
    int tj0 = (rem & 15) * 4;            // 4 col-tiles  (N block of 64)
    int s0 = poff[k], s1 = poff[k + 1];
    v8f acc[8];
#pragma unroll
    for (int t = 0; t < 8; ++t) acc[t] = (v8f){0.f,0.f,0.f,0.f,0.f,0.f,0.f,0.f};
    for (int s = s0; s < s1; s += 4) {
        v2f a0 = load_frag(X, lane, s, (ti0 + 0) * 16, 1024);
        v2f a1 = load_frag(X, lane, s, (ti0 + 1) * 16, 1024);
        v2f b0 = load_frag(X, lane, s, (tj0 + 0) * 16, 1024);
        v2f b1 = load_frag(X, lane, s, (tj0 + 1) * 16, 1024);
        v2f b2 = load_frag(X, lane, s, (tj0 + 2) * 16, 1024);
        v2f b3 = load_frag(X, lane, s, (tj0 + 3) * 16, 1024);
        acc[0] = WMMA_F32(a0, b0, acc[0]);
        acc[1] = WMMA_F32(a0, b1, acc[1]);
        acc[2] = WMMA_F32(a0, b2, acc[2]);
        acc[3] = WMMA_F32(a0, b3, acc[3]);
        acc[4] = WMMA_F32(a1, b0, acc[4]);
        acc[5] = WMMA_F32(a1, b1, acc[5]);
        acc[6] = WMMA_F32(a1, b2, acc[6]);
        acc[7] = WMMA_F32(a1, b3, acc[7]);
    }
    float cw  = cwn[k];
    int n     = lane & 15;
    int mbase = (lane >> 4) * 8;
    float term = 0.0f;
#pragma unroll
    for (int p = 0; p < 2; ++p) {
#pragma unroll
        for (int q = 0; q < 4; ++q) {
            v8f cacc = acc[p * 4 + q];
#pragma unroll
            for (int v = 0; v < 8; ++v) {
                int r  = (ti0 + p) * 16 + mbase + v;  // flattened (i,j)
                int cc = (tj0 + q) * 16 + n;          // flattened (l,m)
                int i = r >> 5,  j  = r & 31;
                int l = cc >> 5, mm = cc & 31;
                if (l != mm) {                        // off_diag on last two indices
                    float w4  = permw4(i, j, l, mm);
                    float t4  = (float)(((i == l) && (j == mm)) + ((i == mm) && (j == l)));
                    float t4c = compressf(t4, C4A, P4, C4S);
                    float m4c = compressf(cacc[v], C4A, P4, C4S);
                    float d = m4c - t4c;
                    term += 0.125f * cw * w4 * d * d;
                }
            }
        }
    }
    for (int m = 16; m > 0; m >>= 1) term += __shfl_xor(term, m, 32);
    if (lane == 0) part[wave] = term;
}

// deterministic final reduction of all partial sums
__global__ void k_reduce(const float* __restrict__ part, int n, float* __restrict__ out) {
    __shared__ float sh[256];
    float a = 0.0f;
    for (int i = threadIdx.x; i < n; i += 256) a += part[i];
    sh[threadIdx.x] = a;
    __syncthreads();
    for (int s = 128; s > 0; s >>= 1) {
        if (threadIdx.x < s) sh[threadIdx.x] += sh[threadIdx.x + s];
        __syncthreads();
    }
    if (threadIdx.x == 0) out[0] = sh[0];
}

extern "C" void kernel_launch(void* const* d_in, const int* in_sizes, int n_in,
                              void* d_out, int out_size, void* d_ws, size_t ws_size,
                              hipStream_t stream) {
    (void)in_sizes; (void)n_in; (void)out_size; (void)ws_size;
    const float* emb    = (const float*)d_in[0];   // (B, 32)
    const float* cen    = (const float*)d_in[1];   // (K, 32)
    const float* logits = (const float*)d_in[2];   // (B, K)
    float* out = (float*)d_out;

    // workspace layout
    float* X   = (float*)d_ws;                       // PT*1024
    float* Dm  = X + (size_t)PT * 1024;              // PT*32
    float* cwn = Dm + (size_t)PT * DIMC;             // KC
    float* cnt = cwn + KC;                           // KC
    float* part = cnt + KC;                          // NP4 + NP3 + NP12
    int* asg   = (int*)(part + (NP4 + NP3 + NP12));  // BC
    int* order = asg + BC;                           // PT
    int* poff  = order + PT;                         // KC+1

    const int nPart = NP4 + NP3 + NP12;

    k_init<<<((size_t)PT * 1024 + 255) / 256, 256, 0, stream>>>(X, Dm, order);
    k_assign<<<(BC + 255) / 256, 256, 0, stream>>>(logits, asg);
    k_meta<<<1, 32, 0, stream>>>(asg, poff, cwn, cnt);
    k_scatter<<<(BC + 255) / 256, 256, 0, stream>>>(asg, poff, order);
    k_build<<<PT, 256, 0, stream>>>(emb, cen, order, poff, X, Dm);
    k_m12<<<(KC * 1024) / 256, 256, 0, stream>>>(X, Dm, poff, cwn, cnt, part);
    k_m3<<<(NP3 * 32) / 256, 256, 0, stream>>>(X, Dm, poff, cwn, part);
    k_m4<<<(NP4 * 32) / 256, 256, 0, stream>>>(X, poff, cwn, part);
    k_reduce<<<1, 256, 0, stream>>>(part, nPart, out);
}